// AffinityLoss_17918603559592
// MI455X (gfx1250) — compile-verified
//
#include <hip/hip_runtime.h>
#include <hip/hip_bf16.h>

// ---------------- problem constants (fixed by the reference) ----------------
#define NUM_CLASSES 182
#define IGNORE_LBL  255
#define B_DIM       4
#define N_DIM       4096                 // (512/8)*(512/8)
#define NN          (N_DIM * N_DIM)      // 2^24 elements per batch
#define TOT_ELEMS   (4LL * (long long)NN)// 67,108,864
#define MAIN_BLOCKS 2048
#define THREADS     256

typedef float v2f __attribute__((ext_vector_type(2)));
typedef float v4f __attribute__((ext_vector_type(4)));
typedef float v8f __attribute__((ext_vector_type(8)));
typedef int   v4i __attribute__((ext_vector_type(4)));

// ---------------------------------------------------------------------------
// Kernel 1: labels[b*4096 + i] = (target[b, 8*y, 8*x] == 255) ? 182 : value
// 16384 work-items total; output table is 64 KB and stays hot in cache.
// ---------------------------------------------------------------------------
__global__ void build_labels_kernel(const int* __restrict__ target,
                                    int* __restrict__ labels) {
  int idx = blockIdx.x * blockDim.x + threadIdx.x;   // [0, 16384)
  int b   = idx >> 12;
  int rem = idx & 4095;
  int y   = rem >> 6;
  int x   = rem & 63;
  // target is (4, 512, 512); sample with stride 8 in both spatial dims
  int v = target[(b << 18) + (y << 12) + (x << 3)];
  labels[idx] = (v == IGNORE_LBL) ? NUM_CLASSES : v;
}

// ---------------------------------------------------------------------------
// Kernel 2: streaming loss + reduction to per-block partials.
// Bandwidth-bound: one NT read of 256 MB of f32. Each thread does exactly 32
// float4 iterations (uniform -> EXEC all-ones at the WMMA reduction).
// ---------------------------------------------------------------------------
__global__ void affinity_loss_main_kernel(const float* __restrict__ pred,
                                          const int* __restrict__ labels,
                                          float* __restrict__ partials) {
  __shared__ float wave_sums[THREADS / 32];

  const long long v4count = TOT_ELEMS >> 2;                 // 16,777,216 vec4
  const long long tid     = (long long)blockIdx.x * blockDim.x + threadIdx.x;
  const long long stride  = (long long)MAIN_BLOCKS * THREADS;

  const v4f* __restrict__ p4 = (const v4f*)pred;
  float acc = 0.0f;

  for (long long v = tid; v < v4count; v += stride) {
    long long e = v << 2;                       // flat element index
    int b   = (int)(e >> 24);                   // NN == 2^24
    int rem = (int)(e & (long long)(NN - 1));
    int i   = rem >> 12;                        // row
    int j   = rem & (N_DIM - 1);                // col (multiple of 4)

    const int* labRow = labels + (b << 12);
    int li = labRow[i];
    v4i lj = *(const v4i*)(labRow + j);         // 16B aligned (j % 4 == 0)

    v4f p = __builtin_nontemporal_load(p4 + v); // stream, don't pollute L2

    // loss = aff ? -clip(log p, -100) : -clip(log(1-p), -100)
    #pragma unroll
    for (int k = 0; k < 4; ++k) {
      float pk = p[k];
      float lp = fmaxf(__logf(pk),        -100.0f);
      float lq = fmaxf(__logf(1.0f - pk), -100.0f);
      acc -= (li == lj[k]) ? lp : lq;
    }
  }

  // ---- wave32 reduction via V_WMMA_F32_16X16X4_F32 ------------------------
  // A (16x4 f32): lanes 0-15 supply A[m][0] (VGPR0) / A[m][1] (VGPR1),
  //               lanes 16-31 supply A[m][2] / A[m][3].
  // Put acc in K-slot 0/2, zero in 1/3; B = all ones (every element == 1).
  // Then D[m][n] = s_m + s_{m+16} for every n, so each lane's 8 C-VGPRs sum
  // to half the wave total; one shfl_xor(16) completes the wave sum.
  v2f a;  a[0] = acc;  a[1] = 0.0f;
  v2f bo; bo[0] = 1.0f; bo[1] = 1.0f;
  v8f c = {0.0f, 0.0f, 0.0f, 0.0f, 0.0f, 0.0f, 0.0f, 0.0f};
  c = __builtin_amdgcn_wmma_f32_16x16x4_f32(
        /*neg_a=*/false, a, /*neg_b=*/false, bo,
        /*c_mod=*/(short)0, c, /*reuse_a=*/false, /*reuse_b=*/false);

  float t = ((c[0] + c[1]) + (c[2] + c[3])) + ((c[4] + c[5]) + (c[6] + c[7]));
  t += __shfl_xor(t, 16, 32);                   // add lanes16-31 half

  int lane = threadIdx.x & 31;
  int wave = threadIdx.x >> 5;
  if (lane == 0) wave_sums[wave] = t;
  __syncthreads();
  if (threadIdx.x == 0) {
    float s = 0.0f;
    #pragma unroll
    for (int w = 0; w < THREADS / 32; ++w) s += wave_sums[w];
    partials[blockIdx.x] = s;
  }
}

// ---------------------------------------------------------------------------
// Kernel 3: deterministic final reduction (single block, double accumulate).
// ---------------------------------------------------------------------------
__global__ void final_reduce_kernel(const float* __restrict__ partials,
                                    float* __restrict__ out) {
  __shared__ double sd[THREADS];
  double s = 0.0;
  for (int idx = threadIdx.x; idx < MAIN_BLOCKS; idx += blockDim.x)
    s += (double)partials[idx];
  sd[threadIdx.x] = s;
  __syncthreads();
  for (int off = THREADS / 2; off > 0; off >>= 1) {
    if (threadIdx.x < off) sd[threadIdx.x] += sd[threadIdx.x + off];
    __syncthreads();
  }
  if (threadIdx.x == 0)
    out[0] = (float)(sd[0] / (double)TOT_ELEMS);
}

// ---------------------------------------------------------------------------
extern "C" void kernel_launch(void* const* d_in, const int* in_sizes, int n_in,
                              void* d_out, int out_size, void* d_ws, size_t ws_size,
                              hipStream_t stream) {
  const float* pred   = (const float*)d_in[0];   // (4, 4096, 4096) f32
  const int*   target = (const int*)d_in[1];     // (4, 512, 512) int

  int*   labels   = (int*)d_ws;                                  // 64 KB
  float* partials = (float*)((char*)d_ws + B_DIM * N_DIM * 4);   // 8 KB

  build_labels_kernel<<<(B_DIM * N_DIM) / THREADS, THREADS, 0, stream>>>(
      target, labels);

  affinity_loss_main_kernel<<<MAIN_BLOCKS, THREADS, 0, stream>>>(
      pred, labels, partials);

  final_reduce_kernel<<<1, THREADS, 0, stream>>>(partials, (float*)d_out);
}